// S_Interp_58506044506513
// MI455X (gfx1250) — compile-verified
//
#include <hip/hip_runtime.h>
#include <math.h>

namespace {

constexpr int Bn = 8, Tn = 512, Dn = 128, On = 128;

typedef __attribute__((ext_vector_type(2))) float v2f;
typedef __attribute__((ext_vector_type(8))) float v8f;

// One wave per (b, d). Block = 256 threads = 8 wave32 -> 8 consecutive d's.
// Grid = B * D/8 = 128 blocks.
//
// Per (b,d):  W[o]  = sum_t K[t,o]    * m[t]        (K = exp(-alpha*(ts[t]-ref[o])^2))
//             Y[o]  = sum_t K[t,o]    * m[t]*x[t]
//             Yt[o] = sum_t K[t,o]^10 * m[t]*x[t]
// out[b,d,o] = Y/W ; out[b,D+d,o] = ln(W) ; out[b,2D+d,o] = Yt/W
//
// t-reduction via V_WMMA_F32_16X16X4_F32:
//   A (16x4): row0 = m[t..t+3], row1 = (m*x)[t..t+3], rows 2..15 = 0
//   B (4x16): K[t..t+3, o_tile]   (second WMMA: K^10)
//   D (16x16 f32) accumulates over 128 t-chunks; rows 0/1 hold W/Y.
__global__ __launch_bounds__(256) void s_interp_wmma(
    const float* __restrict__ x_ts,    // [B,T,D]
    const float* __restrict__ x_mask,  // [B,T,D]
    const float* __restrict__ ts_tt,   // [B,T]
    const float* __restrict__ kern,    // [D]
    float* __restrict__ out)           // [B,3D,O]
{
  __shared__ float lds_ts[Tn];
  __shared__ float lds_am[8][Tn];   // m = mask + 1e-12
  __shared__ float lds_ax[8][Tn];   // m * x

  const int b    = (int)blockIdx.x >> 4;   // 16 blocks per batch
  const int dgrp = (int)blockIdx.x & 15;   // which group of 8 d's
  const int tid  = (int)threadIdx.x;
  const int wave = tid >> 5;
  const int lane = tid & 31;
  const int d    = dgrp * 8 + wave;

  // Stage ts[b,:] and this block's 8 columns of mask/x into LDS.
  for (int t = tid; t < Tn; t += 256) lds_ts[t] = ts_tt[b * Tn + t];
  for (int i = tid; i < Tn * 8; i += 256) {
    const int t  = i >> 3;
    const int j  = i & 7;
    const size_t g = (size_t)(b * Tn + t) * Dn + (dgrp * 8 + j);
    // Prefetch the next stripe while this one is in flight (global_prefetch_b8).
    __builtin_prefetch(&x_ts[g + 32 * Dn], 0, 1);
    __builtin_prefetch(&x_mask[g + 32 * Dn], 0, 1);
    const float m  = x_mask[g] + 1e-12f;
    const float xv = x_ts[g];
    lds_am[j][t] = m;
    lds_ax[j][t] = m * xv;
  }
  __syncthreads();

  // alpha_d = softplus(kernel[d]); fold ln->log2 conversion into the exp2 arg.
  const float kv    = kern[d];
  const float alpha = (kv > 20.0f) ? kv : log1pf(expf(kv));
  const float c1    = -alpha * 1.4426950408889634f;  // -alpha * log2(e)

  // Lane roles per ISA f32 WMMA layouts:
  //  A 16x4 : lanes 0-15 = rows M (K=0,1 in the 2 VGPRs), lanes 16-31 = K=2,3.
  //  B 4x16 : lanes 0-15 = cols N (rows 0,1), lanes 16-31 = rows 2,3.
  // -> every lane works on t-pair (tA, tA+1) with tA = tc + 2*(lane>=16).
  const int   mrow   = lane & 15;
  const int   thalf  = (lane >> 4) << 1;
  const float ro_scl = 1.0f / (float)(On - 1);

  // Branchless per-lane row selectors (hoisted; keeps the hot loop EXEC-clean).
  const float s_am = (mrow == 0) ? 1.0f : 0.0f;   // A row 0 carries m
  const float s_ax = (mrow == 1) ? 1.0f : 0.0f;   // A row 1 carries m*x

  const float* __restrict__ amp = &lds_am[wave][0];
  const float* __restrict__ axp = &lds_ax[wave][0];

  for (int ot = 0; ot < 8; ++ot) {
    const int   o  = ot * 16 + mrow;
    const float ro = (float)o * ro_scl;
    v8f acc1 = {0.f, 0.f, 0.f, 0.f, 0.f, 0.f, 0.f, 0.f};
    v8f acc2 = {0.f, 0.f, 0.f, 0.f, 0.f, 0.f, 0.f, 0.f};

    for (int tc = 0; tc < Tn; tc += 4) {
      const int tA = tc + thalf;                     // even -> 8B aligned
      const float2 tsp = *(const float2*)&lds_ts[tA];  // ds_load_b64
      const float2 amv = *(const float2*)&amp[tA];     // ds_load_b64
      const float2 axv = *(const float2*)&axp[tA];     // ds_load_b64

      const float dA = tsp.x - ro;
      const float dB = tsp.y - ro;
      // K = exp2(c1 * norm); K^10 via square-chain (no 2nd transcendental)
      const float kA = __builtin_amdgcn_exp2f(c1 * dA * dA);
      const float kB = __builtin_amdgcn_exp2f(c1 * dB * dB);
      const float p2A = kA * kA, p4A = p2A * p2A;
      const float p2B = kB * kB, p4B = p2B * p2B;
      const float kA10 = p4A * p4A * p2A;
      const float kB10 = p4B * p4B * p2B;

      v2f Bm, B10, A1, A2;
      Bm.x  = kA;   Bm.y  = kB;
      B10.x = kA10; B10.y = kB10;
      // Branchless FMA row-select: no cndmask chains, no saveexec.
      A1.x = fmaf(s_am, amv.x, s_ax * axv.x);
      A1.y = fmaf(s_am, amv.y, s_ax * axv.y);
      A2.x = s_am * axv.x;
      A2.y = s_am * axv.y;

      acc1 = __builtin_amdgcn_wmma_f32_16x16x4_f32(false, A1, false, Bm,
                                                   (short)0, acc1, false, false);
      acc2 = __builtin_amdgcn_wmma_f32_16x16x4_f32(false, A2, false, B10,
                                                   (short)0, acc2, false, false);
    }

    // C/D layout: VGPR0 = row 0 (lanes 0-15), VGPR1 = row 1 (lanes 0-15).
    if (lane < 16) {
      const float W    = acc1[0];
      const float Y    = acc1[1];
      const float Yt   = acc2[0];
      const float invW = 1.0f / W;
      float* ob = out + (size_t)b * (3 * Dn * On);
      ob[(size_t)d * On + o]            = Y * invW;
      ob[(size_t)(Dn + d) * On + o]     = __builtin_amdgcn_logf(W) * 0.69314718055994531f;
      ob[(size_t)(2 * Dn + d) * On + o] = Yt * invW;
    }
  }
}

}  // namespace

extern "C" void kernel_launch(void* const* d_in, const int* in_sizes, int n_in,
                              void* d_out, int out_size, void* d_ws, size_t ws_size,
                              hipStream_t stream) {
  (void)in_sizes; (void)n_in; (void)out_size; (void)d_ws; (void)ws_size;
  const float* x_ts   = (const float*)d_in[0];
  const float* x_mask = (const float*)d_in[1];
  const float* ts_tt  = (const float*)d_in[2];
  // d_in[3] = rec_mask (unused in non-reconstruction path)
  const float* kern   = (const float*)d_in[4];
  // d_in[5] = tt_max (compile-time constant O=128)
  float* out = (float*)d_out;

  dim3 grid(Bn * (Dn / 8));  // 128 blocks: one batch-slice of 8 d's each
  dim3 block(256);           // 8 wave32's, one d per wave
  hipLaunchKernelGGL(s_interp_wmma, grid, block, 0, stream,
                     x_ts, x_mask, ts_tt, kern, out);
}